// OverlapAdd_19490561590070
// MI455X (gfx1250) — compile-verified
//
#include <hip/hip_runtime.h>
#include <stdint.h>

// OverlapAdd: y[b,c,t] = sum_{k=0..3} x[b, k*128+c, t-k]  (zero-padded at t<k)
// B=8, C_in=512, C_out=128, T=16000, f32. Pure HBM-bound (~328 MB, ~14us @ 23.3TB/s).
// Strategy: TDM (tensor_load_to_lds) stages the 4 strided rows into LDS with the
// k-shift PRE-APPLIED (row k loaded from t0-k), so compute is 4 aligned
// ds_load_b128 + packed f32 adds + one aligned global_store_b128 per float4.

#define T_LEN   16000
#define C_IN    512
#define C_OUT   128
#define B_DIM   8
#define TILE    2000
#define N_TILES (T_LEN / TILE)   // 8, exact

typedef __attribute__((ext_vector_type(4))) unsigned int u32x4;
typedef __attribute__((ext_vector_type(4))) int          i32x4;
typedef __attribute__((ext_vector_type(8))) int          i32x8;

// Issue one 1-row TDM load: copies nelem floats from global gptr to the LDS byte address lds_byte
__device__ __forceinline__ void tdm_load_row(const float* gptr, unsigned lds_byte,
                                             unsigned nelem)
{
    unsigned long long ga = (unsigned long long)(uintptr_t)gptr;

    // D# group 0 (128b): count=1 | lds_addr | global_addr[56:0] | type=2
    u32x4 g0;
    g0[0] = 1u;                                      // count=1 (valid), user mode
    g0[1] = lds_byte;                                // lds_addr (bytes)
    g0[2] = (unsigned)(ga & 0xffffffffull);          // global_addr[31:0]
    g0[3] = (unsigned)((ga >> 32) & 0x1ffffffull)    // global_addr[56:32]
          | (2u << 30);                              // type = 2 ("image")

    // D# group 1 (256b): wg_mask=0, data_size=2 (4B), dims/strides
    i32x8 g1;
    g1[0] = (int)(2u << 16);                         // data_size = 4 bytes
    g1[1] = (int)((nelem & 0xffffu) << 16);          // tensor_dim0[15:0]
    g1[2] = (int)((nelem >> 16) & 0xffffu)           // tensor_dim0[31:16]
          | (1  << 16);                              // tensor_dim1 = 1
    g1[3] = (int)((nelem & 0xffffu) << 16);          // tile_dim0 = nelem
    g1[4] = 1;                                       // tile_dim1 = 1, tile_dim2 = 0
    g1[5] = 0;                                       // tensor_dim0_stride (1 row)
    g1[6] = 0;
    g1[7] = 0;

    i32x4 gz4 = {0, 0, 0, 0};                        // groups 2/3: <=2D tensor
    i32x8 gz8 = {0, 0, 0, 0, 0, 0, 0, 0};            // trailing group (clang-23 form)

    __builtin_amdgcn_tensor_load_to_lds(g0, g1, gz4, gz4, gz8, 0);
}

__global__ __launch_bounds__(256)
void overlap_add_tdm_kernel(const float* __restrict__ x, float* __restrict__ y)
{
    // lds[k][j] == x[b, k*128+c, t0 + j - k]   (zero where t0+j-k < 0)
    __shared__ __align__(16) float lds[4][TILE];     // 32,000 B

    const int tid  = threadIdx.x;
    const int wg   = blockIdx.x;
    const int tile = wg & (N_TILES - 1);
    const int bc   = wg >> 3;                        // log2(N_TILES) = 3
    const int c    = bc & (C_OUT - 1);
    const int b    = bc >> 7;                        // log2(C_OUT) = 7
    const int t0   = tile * TILE;

    // Wave 0 drives the TDM: row k sourced from global index t0-k so the
    // k-shift is baked into the LDS image (all compute reads 16B-aligned).
    if (tid < 32) {
        const bool first = (tile == 0);
        for (int k = 0; k < 4; ++k) {
            const int start  = first ? 0 : (t0 - k); // avoid t<0 reads on tile 0
            const unsigned skip = first ? (unsigned)k : 0u;   // dwords shifted in LDS
            const float* row = x
                + (size_t)(b * C_IN + k * C_OUT + c) * T_LEN + start;
            unsigned lds_byte = (unsigned)(uintptr_t)(&lds[k][0]) + skip * 4u;
            tdm_load_row(row, lds_byte, (unsigned)TILE - skip);
        }
        __builtin_amdgcn_s_wait_tensorcnt(0);        // all 4 row tiles resident

        if (first && tid < 16) {                     // zero the t<0 halo entries
            const int k = tid >> 2, i = tid & 3;
            if (i < k) lds[k][i] = 0.0f;
        }
    }
    __syncthreads();

    // y[t0+j .. +3] = lds0[j..] + lds1[j..] + lds2[j..] + lds3[j..]
    float* orow = y + (size_t)(b * C_OUT + c) * T_LEN + t0;
    for (int j4 = tid; j4 < TILE / 4; j4 += 256) {
        const int j = j4 * 4;
        const float4 a0 = *reinterpret_cast<const float4*>(&lds[0][j]);
        const float4 a1 = *reinterpret_cast<const float4*>(&lds[1][j]);
        const float4 a2 = *reinterpret_cast<const float4*>(&lds[2][j]);
        const float4 a3 = *reinterpret_cast<const float4*>(&lds[3][j]);
        float4 o;
        o.x = (a0.x + a1.x) + (a2.x + a3.x);
        o.y = (a0.y + a1.y) + (a2.y + a3.y);
        o.z = (a0.z + a1.z) + (a2.z + a3.z);
        o.w = (a0.w + a1.w) + (a2.w + a3.w);
        *reinterpret_cast<float4*>(orow + j) = o;
    }
}

extern "C" void kernel_launch(void* const* d_in, const int* in_sizes, int n_in,
                              void* d_out, int out_size, void* d_ws, size_t ws_size,
                              hipStream_t stream)
{
    const float* x = (const float*)d_in[0];
    float*       y = (float*)d_out;

    dim3 grid(B_DIM * C_OUT * N_TILES);              // 8192 workgroups
    dim3 block(256);                                 // 8 waves (wave32)
    overlap_add_tdm_kernel<<<grid, block, 0, stream>>>(x, y);
}